// FIRFilterHardware_29214367547585
// MI455X (gfx1250) — compile-verified
//
#include <hip/hip_runtime.h>

typedef __attribute__((ext_vector_type(2))) float v2f;
typedef __attribute__((ext_vector_type(4))) float v4f;
typedef __attribute__((ext_vector_type(8))) float v8f;

#define TAPS 32
#define WAVE_OUT 256                             // 16x16 output tile per wave
#define WAVES_PER_BLOCK 8
#define BLOCK_OUT (WAVE_OUT * WAVES_PER_BLOCK)   // 2048 outputs per block
#define LDS_X (BLOCK_OUT + 64)                   // 2112 floats = 528 float4

__global__ __launch_bounds__(256)
void fir_wmma_f32_kernel(const float* __restrict__ x,
                         const float* __restrict__ taps,
                         float* __restrict__ y,
                         int n) {
    __shared__ __attribute__((aligned(16))) float xs[LDS_X];
    __shared__ float tp[TAPS];

    const int tid = threadIdx.x;
    const int blockStart = (int)blockIdx.x * BLOCK_OUT;
    const int g0 = blockStart - 32;              // 128B-aligned for every block
    // Edge blocks: block 0 (negative window + n<32 masking) and any block whose
    // window [g0, g0+2112) or outputs run past n.
    const bool edge = (blockIdx.x == 0) | (blockStart + BLOCK_OUT + 32 > n);

    if (!edge) {
        // Fast path: aligned 128-bit global loads -> 128-bit LDS stores.
        const v4f* __restrict__ xv = (const v4f*)(x + g0);
        v4f* sv = (v4f*)xs;
        sv[tid]       = xv[tid];
        sv[tid + 256] = xv[tid + 256];
        if (tid < 16) sv[tid + 512] = xv[tid + 512];
    } else {
        for (int t = tid; t < LDS_X; t += 256) {
            int gi = g0 + t;
            xs[t] = (gi >= 0 && gi < n) ? x[gi] : 0.0f;
        }
    }
    if (tid < TAPS) tp[tid] = taps[tid];
    __syncthreads();

    const int wave = tid >> 5;
    const int lane = tid & 31;
    const int half = lane >> 4;   // 0: lanes 0-15, 1: lanes 16-31
    const int j    = lane & 15;   // B/D column; also A row (same expression)

    // Banded-Toeplitz B operand per K-step (constant per lane, held in VGPRs).
    // B[p][j] = taps[j + 31 - p] if 0 <= j+31-p < 32 else 0.
    // 4x16 B lane layout: this lane holds rows p0 + 2*half and p0 + 2*half + 1.
    v2f b[12];
#pragma unroll
    for (int s = 0; s < 12; ++s) {
        int p0 = 4 * s + 2 * half;
        int k0 = j + 31 - p0;
        int k1 = k0 - 1;
        b[s].x = (k0 >= 0 && k0 < TAPS) ? tp[k0] : 0.0f;
        b[s].y = (k1 >= 0 && k1 < TAPS) ? tp[k1] : 0.0f;
    }

    // A[m][p] = xs[256*wave + 16*m + p + 1]   (m = row = lane&15 = j)
    const int abase = 256 * wave + 16 * j + 1;

    v8f acc = {};
#pragma unroll
    for (int s = 0; s < 12; ++s) {
        int p0 = 4 * s + 2 * half;
        v2f a;
        a.x = xs[abase + p0];
        a.y = xs[abase + p0 + 1];
        // D = A(16x4 f32) * B(4x16 f32) + C ; exact in f32 for this data.
        acc = __builtin_amdgcn_wmma_f32_16x16x4_f32(
            /*neg_a=*/false, a, /*neg_b=*/false, b[s],
            /*c_mod=*/(short)0, acc, /*reuse_a=*/false, /*reuse_b=*/false);
    }

    // Store: VGPR v holds row M=v (lanes 0-15) / M=v+8 (lanes 16-31), col = j.
    // Each (v, half) writes one contiguous 64B segment -> fully coalesced.
    const int base = blockStart + wave * WAVE_OUT + 16 * 8 * half + j;
    const float inv_scale = 1.0f / 65536.0f;     // 2^-16, exact
    if (!edge) {
#pragma unroll
        for (int v = 0; v < 8; ++v) {
            y[base + 16 * v] = floorf(acc[v] * inv_scale);
        }
    } else {
#pragma unroll
        for (int v = 0; v < 8; ++v) {
            int nIdx = base + 16 * v;
            if (nIdx < n) {
                y[nIdx] = (nIdx < TAPS) ? 0.0f : floorf(acc[v] * inv_scale);
            }
        }
    }
}

extern "C" void kernel_launch(void* const* d_in, const int* in_sizes, int n_in,
                              void* d_out, int out_size, void* d_ws, size_t ws_size,
                              hipStream_t stream) {
    const float* x    = (const float*)d_in[0];
    const float* taps = (const float*)d_in[1];
    float* y          = (float*)d_out;
    int n = in_sizes[0];

    int blocks = (n + BLOCK_OUT - 1) / BLOCK_OUT;
    fir_wmma_f32_kernel<<<blocks, 256, 0, stream>>>(x, taps, y, n);
}